// LinearQuantLin_66056597012844
// MI455X (gfx1250) — compile-verified
//
#include <hip/hip_runtime.h>

// D = A(16x4 f32) * B(4x16 f32) + C(16x16 f32), wave32 WMMA
typedef __attribute__((ext_vector_type(2))) float v2f;
typedef __attribute__((ext_vector_type(8))) float v8f;

#define IN_F   8192
#define OUT_F  8192
#define BATCH  16
#define WAVES  8
#define KSLICE (IN_F / WAVES)   // 1024 K per wave

__global__ __launch_bounds__(256) void qlin_fwd_gemm(
    const float* __restrict__ x,      // [16, 8192]
    const float* __restrict__ w,      // [8192, 8192] (out-major)
    const float* __restrict__ bias,   // [8192]
    float* __restrict__ out)          // [16, 8192]
{
    __shared__ float red[WAVES * 256];   // 8 partial 16x16 f32 tiles

    const int tid  = threadIdx.x;
    const int wave = tid >> 5;
    const int lane = tid & 31;
    const int n0   = blockIdx.x * 16;    // 16 outputs per block

    // WMMA f32 16x16x4 fragment mapping (wave32):
    //   A: lane holds row m = lane&15, K = kbase + (lane>>4)*2 + {0,1}
    //   B: lane holds col n = lane&15, K = kbase + (lane>>4)*2 + {0,1}
    const int m     = lane & 15;
    const int khalf = (lane >> 4) * 2;
    const int k0    = wave * KSLICE;

    const float* xfrag = x + m * IN_F + k0 + khalf;            // A = x
    const float* wfrag = w + (n0 + m) * IN_F + k0 + khalf;     // B[k][n] = W[n][k]

    v8f acc = {};
#pragma unroll 8
    for (int kk = 0; kk < KSLICE; kk += 4) {
        v2f a;
        a.x = xfrag[kk + 0];
        a.y = xfrag[kk + 1];
        v2f b;
        // W is streamed exactly once across the whole grid: non-temporal so it
        // doesn't evict the hot 512KB x tile from L2.
        b.x = __builtin_nontemporal_load(wfrag + kk + 0);
        b.y = __builtin_nontemporal_load(wfrag + kk + 1);
        // 8 args: (neg_a, A, neg_b, B, c_mod, C, reuse_a, reuse_b)
        acc = __builtin_amdgcn_wmma_f32_16x16x4_f32(
            false, a, false, b, (short)0, acc, false, false);
    }

    // Stash this wave's partial 16x16 tile: element index e = r*32 + lane
#pragma unroll
    for (int r = 0; r < 8; ++r)
        red[wave * 256 + r * 32 + lane] = acc[r];
    __syncthreads();

    // 256 threads: each owns one of the 256 tile elements; fixed-order sum
    // across the 8 K-slices (deterministic), then bias add + store.
    float s = 0.f;
#pragma unroll
    for (int wv = 0; wv < WAVES; ++wv)
        s += red[wv * 256 + tid];

    const int r  = tid >> 5;          // accumulator VGPR index 0..7
    const int el = tid & 31;          // lane within tile row-pair
    const int om = r + ((el >> 4) << 3);   // C/D layout: lanes 16-31 are M+8
    const int on = n0 + (el & 15);
    out[om * OUT_F + on] = s + bias[on];
}

extern "C" void kernel_launch(void* const* d_in, const int* in_sizes, int n_in,
                              void* d_out, int out_size, void* d_ws, size_t ws_size,
                              hipStream_t stream) {
    const float* x    = (const float*)d_in[0];
    const float* w    = (const float*)d_in[1];
    const float* bias = (const float*)d_in[2];
    // d_in[3] = alpha: unused in forward (quantization enters backward only)
    float* out = (float*)d_out;

    dim3 grid(OUT_F / 16);   // 512 blocks, one 16-wide output tile each
    dim3 block(256);         // 8 waves, split-K x8 per block
    qlin_fwd_gemm<<<grid, block, 0, stream>>>(x, w, bias, out);
}